// EntmaxBisect_44298292691123
// MI455X (gfx1250) — compile-verified
//
#include <hip/hip_runtime.h>

// Entmax-1.5 bisection, rows of 4096 fp32 (32768 rows).
// alpha = 1.5 -> exp = 1/(alpha-1) = 2 (pure squaring), (1/d)^0.5 = 1/64.
//
// Design (MI455X / gfx1250, wave32):
//   * ONE wave owns ONE row: 4096/32 = 128 elems/lane = 32 v4f registers
//     (128 VGPRs) that are the DIRECT destinations of 32 clause'd
//     global_load_b128s, scaled in place -> peak ~145 VGPRs, no spills,
//     no s_set_vgpr_msb traffic.
//   * No LDS, no barriers: every reduction is a 5-step shfl_xor butterfly
//     (lane-permute path), result already broadcast to all 32 lanes.
//   * v4f math lowers to packed v_pk_add_f32 / v_pk_fma_f32 (CDNA5 2-wide
//     fp32) + VOPD-pairable v_max_f32; 8 independent packed FMA chains.
//   * Single HBM read + single HBM write per element (roofline floor
//     ~43 us at 23.3 TB/s); the 50-iteration loop is pure register VALU.

typedef float v4f __attribute__((ext_vector_type(4)));

namespace {
constexpr int   kD        = 4096;
constexpr int   kLanes    = 32;
constexpr int   kQuads    = kD / kLanes / 4;   // 32 v4f per lane
constexpr int   kWavesPB  = 8;                 // independent rows per block
constexpr int   kThreads  = kWavesPB * kLanes; // 256
constexpr int   kIters    = 50;
constexpr int   kAcc      = 4;                 // v4f accumulator chains
constexpr float kAlphaM1  = 0.5f;              // alpha - 1
constexpr float kInvDPow  = 0.015625f;         // (1/4096)^(alpha-1) = 1/64
}

__device__ __forceinline__ float waveSum(float v) {
#pragma unroll
    for (int off = 16; off > 0; off >>= 1) v += __shfl_xor(v, off, 32);
    return v;
}

__device__ __forceinline__ float waveMax(float v) {
#pragma unroll
    for (int off = 16; off > 0; off >>= 1) v = fmaxf(v, __shfl_xor(v, off, 32));
    return v;
}

// Row-wide sum of relu(xs - tau)^2, broadcast to all lanes of the wave.
__device__ __forceinline__ float rowSumSq(const v4f* __restrict__ xs, float tau) {
    const v4f tau4  = {tau, tau, tau, tau};
    const v4f zero4 = {0.0f, 0.0f, 0.0f, 0.0f};
    v4f acc[kAcc];
#pragma unroll
    for (int a = 0; a < kAcc; ++a) acc[a] = zero4;
#pragma unroll
    for (int j = 0; j < kQuads; ++j) {
        v4f t = xs[j] - tau4;                       // 2x v_pk_add_f32 (neg)
        t = __builtin_elementwise_max(t, zero4);    // 4x v_max_f32 (VOPD-pairable)
        acc[j & (kAcc - 1)] += t * t;               // 2x v_pk_fma_f32
    }
    acc[0] += acc[2];
    acc[1] += acc[3];
    acc[0] += acc[1];
    return waveSum((acc[0].x + acc[0].y) + (acc[0].z + acc[0].w));
}

__global__ __launch_bounds__(kThreads)
void entmax15_bisect_kernel(const float* __restrict__ X,
                            float* __restrict__ out) {
    const int lane = threadIdx.x & (kLanes - 1);
    const int wid  = threadIdx.x >> 5;
    const int row  = blockIdx.x * kWavesPB + wid;

    const size_t base = (size_t)row * (kD / 4);
    const v4f* __restrict__ X4 = reinterpret_cast<const v4f*>(X);

    // ---- Load the whole row straight into its register home ----
    v4f xs[kQuads];
#pragma unroll
    for (int j = 0; j < kQuads; ++j) {
        xs[j] = X4[base + lane + j * kLanes];        // coalesced global_load_b128
    }
    // Scale in place by (alpha-1) = 0.5 (v_pk_mul_f32, same registers).
    const v4f half4 = {kAlphaM1, kAlphaM1, kAlphaM1, kAlphaM1};
#pragma unroll
    for (int j = 0; j < kQuads; ++j) xs[j] *= half4;

    // ---- Row max ----
    v4f mm4 = xs[0];
#pragma unroll
    for (int j = 1; j < kQuads; ++j) mm4 = __builtin_elementwise_max(mm4, xs[j]);
    const float m = waveMax(fmaxf(fmaxf(mm4.x, mm4.y), fmaxf(mm4.z, mm4.w)));

    // ---- Bisection setup ----
    float tau_lo = m - 1.0f;                   // _gp_inv(1)
    float dm     = (m - kInvDPow) - tau_lo;    // tau_hi - tau_lo
    float tau_m  = tau_lo;
    const float f_lo = rowSumSq(xs, tau_lo) - 1.0f;

    // ---- 50 bisection iterations, all state lane-uniform ----
    for (int it = 0; it < kIters; ++it) {
        dm *= 0.5f;
        tau_m = tau_lo + dm;
        const float f_m = rowSumSq(xs, tau_m) - 1.0f;
        if (f_m * f_lo >= 0.0f) tau_lo = tau_m;
    }

    // ---- Normalize with the LAST tau_m; recompute p_m (saves 128 VGPRs) ----
    const float s   = rowSumSq(xs, tau_m);
    const float inv = 1.0f / s;
    const v4f tau4  = {tau_m, tau_m, tau_m, tau_m};
    const v4f zero4 = {0.0f, 0.0f, 0.0f, 0.0f};
    const v4f inv4  = {inv, inv, inv, inv};

    v4f* __restrict__ O4 = reinterpret_cast<v4f*>(out);
#pragma unroll
    for (int j = 0; j < kQuads; ++j) {
        v4f t = __builtin_elementwise_max(xs[j] - tau4, zero4);
        O4[base + lane + j * kLanes] = (t * t) * inv4;   // coalesced b128 store
    }
}

extern "C" void kernel_launch(void* const* d_in, const int* in_sizes, int n_in,
                              void* d_out, int out_size, void* d_ws, size_t ws_size,
                              hipStream_t stream) {
    (void)n_in; (void)d_ws; (void)ws_size; (void)out_size;
    const float* X = (const float*)d_in[0];
    float* out     = (float*)d_out;
    const int nrows   = in_sizes[0] / kD;        // 32768
    const int nblocks = nrows / kWavesPB;        // 4096
    entmax15_bisect_kernel<<<dim3(nblocks), dim3(kThreads), 0, stream>>>(X, out);
}